// NeighbourAggregation_13365938225895
// MI455X (gfx1250) — compile-verified
//
#include <hip/hip_runtime.h>

typedef float v2f __attribute__((ext_vector_type(2)));
typedef float v8f __attribute__((ext_vector_type(8)));

// Fast hardware transcendentals (single v_sqrt_f32 / v_exp_f32, ~1 ulp).
__device__ __forceinline__ float fast_sqrt(float x) { return __builtin_amdgcn_sqrtf(x); }
__device__ __forceinline__ float fast_exp2(float x) { return __builtin_amdgcn_exp2f(x); }

#define LOG2E 1.44269504088896340736f

// ---------------------------------------------------------------------------
// Kernel 1: per-particle precompute.
//  rec[i] = (px, py, |p|^2, 0)                       (16B record for distance pass)
//  G[i][0..15] = [f0..f3, f0^2..f3^2, 1, 0,...,0]    (B-matrix rows for WMMA GEMM)
// where f = [pos.x, pos.y, vel.x, vel.y]
// ---------------------------------------------------------------------------
__global__ void precompute_kernel(const float* __restrict__ states,
                                  float4* __restrict__ rec,
                                  float* __restrict__ G, int N) {
    int i = blockIdx.x * blockDim.x + threadIdx.x;
    if (i >= N) return;
    const float4* s = (const float4*)(states + (size_t)i * 8);
    float4 a = s[0];
    float4 b = s[1];
    float px = (a.x + a.z) * 0.5f;
    float py = (a.y + a.w) * 0.5f;
    float vx = (b.x + b.z) * 0.5f;
    float vy = (b.y + b.w) * 0.5f;
    float r  = px * px + py * py;
    rec[i] = make_float4(px, py, r, 0.0f);
    float4* g = (float4*)(G + (size_t)i * 16);
    g[0] = make_float4(px, py, vx, vy);
    g[1] = make_float4(px * px, py * py, vx * vx, vy * vy);
    g[2] = make_float4(1.0f, 0.0f, 0.0f, 0.0f);
    g[3] = make_float4(0.0f, 0.0f, 0.0f, 0.0f);
}

// ---------------------------------------------------------------------------
// Kernel 2: group velocity = mean(vel) over all particles (single block).
// ---------------------------------------------------------------------------
__global__ void groupvel_kernel(const float* __restrict__ G,
                                float* __restrict__ gv, int N) {
    __shared__ float sx[256];
    __shared__ float sy[256];
    int t = threadIdx.x;
    float ax = 0.0f, ay = 0.0f;
    for (int j = t; j < N; j += 256) {
        ax += G[(size_t)j * 16 + 2];
        ay += G[(size_t)j * 16 + 3];
    }
    sx[t] = ax; sy[t] = ay;
    __syncthreads();
    for (int s = 128; s > 0; s >>= 1) {
        if (t < s) { sx[t] += sx[t + s]; sy[t] += sy[t + s]; }
        __syncthreads();
    }
    if (t == 0) {
        gv[0] = sx[0] / (float)N;
        gv[1] = sy[0] / (float)N;
    }
}

// ---------------------------------------------------------------------------
// Kernel 3: all-pairs aggregation.
// 256 threads = 8 waves per block; each wave owns a 16-row i-tile.
// Pass 1: per-row min d^2 over j (row-max logit for softmax stability).
// Pass 2: weights computed directly in WMMA A-layout (lane = M + 16*(K>=2),
//         reg = K&1), accumulated with v_wmma_f32_16x16x4_f32 against
//         B rows [f, f^2, 1, pad...].
// C columns: 0..3 = sum w*f_j, 4..7 = sum w*f_j^2, 8 = sum w.
// ---------------------------------------------------------------------------
__global__ __launch_bounds__(256) void neighbour_kernel(
    const float4* __restrict__ rec,
    const float*  __restrict__ G,
    const float*  __restrict__ gvp,
    const float*  __restrict__ log_tau,
    float*        __restrict__ out, int N)
{
    __shared__ float cbuf[8][256];

    const int tid  = threadIdx.x;
    const int wave = tid >> 5;
    const int lane = tid & 31;
    const int row  = lane & 15;   // local i row (A-matrix M index)
    const int h    = lane >> 4;   // lane half (selects K pairs {0,1} vs {2,3})
    const int i0   = (blockIdx.x * 8 + wave) * 16;
    const int i    = i0 + row;

    const float invtau = 1.0f / __expf(log_tau[0]);

    const float4 ri  = rec[i];
    const float px_i = ri.x, py_i = ri.y, r_i = ri.z;

    // ---- Pass 1: qmin = min_{j != i} (r_i + r_j - 2 p_i.p_j) ----
    float qmin = 3.0e38f;
    #pragma unroll 4
    for (int j = h; j < N; j += 2) {
        float4 rj = rec[j];
        float dot = px_i * rj.x + py_i * rj.y;
        float q   = r_i + rj.z - 2.0f * dot;
        q = (j == i) ? 3.0e38f : q;
        qmin = fminf(qmin, q);
    }
    qmin = fminf(qmin, __shfl_xor(qmin, 16, 32));
    const float mdist = fast_sqrt(fmaxf(qmin, 0.0f) + 1e-8f);  // min dist over j != i

    // Fold exp((mdist - dist)/tau) into exp2(fma(-dist, k2, b2)):
    const float k2 = invtau * LOG2E;
    const float b2 = mdist * k2;

    // ---- Pass 2: WMMA-accumulated weighted sums ----
    v8f acc = {0.0f, 0.0f, 0.0f, 0.0f, 0.0f, 0.0f, 0.0f, 0.0f};

    for (int jt = 0; jt < N; jt += 16) {
        v2f A[4];
        v2f B[4];
        #pragma unroll
        for (int g = 0; g < 4; ++g) {
            #pragma unroll
            for (int p = 0; p < 2; ++p) {
                const int j = jt + 4 * g + p + 2 * h;   // K index -> global j
                float4 rj = rec[j];
                float dot  = px_i * rj.x + py_i * rj.y;
                float q    = r_i + rj.z - 2.0f * dot;
                float dist = fast_sqrt(fmaxf(q, 0.0f) + 1e-8f);
                float w    = fast_exp2(fmaf(-dist, k2, b2));
                w = (j == i) ? 0.0f : w;                // diagonal masked out
                A[g][p] = w;
                B[g][p] = G[(size_t)j * 16 + row];      // B row K, col = lane%16
            }
        }
        #pragma unroll
        for (int g = 0; g < 4; ++g) {
            acc = __builtin_amdgcn_wmma_f32_16x16x4_f32(
                /*neg_a=*/false, A[g], /*neg_b=*/false, B[g],
                /*c_mod=*/(short)0, acc, /*reuse_a=*/false, /*reuse_b=*/false);
        }
    }

    // ---- Spill C (16x16) to LDS: element (M, Ncol) at lane Ncol+16*(M/8), vgpr M%8 ----
    #pragma unroll
    for (int v = 0; v < 8; ++v) {
        cbuf[wave][(v + 8 * h) * 16 + row] = acc[v];
    }
    __syncthreads();

    // ---- Finalize: lanes 0..15 each produce one output row ----
    if (lane < 16) {
        const int ii = i0 + lane;
        const float* c = &cbuf[wave][lane * 16];
        const float S    = c[8];
        const float invS = 1.0f / S;
        const float4 fi4 = *(const float4*)(G + (size_t)ii * 16);
        const float f[4] = {fi4.x, fi4.y, fi4.z, fi4.w};
        const float gvx = gvp[0];
        const float gvy = gvp[1];

        float mu[4], sg[4];
        #pragma unroll
        for (int d = 0; d < 4; ++d) {
            float T = c[d] * invS;          // E[f_j[d]]
            float U = c[4 + d] * invS;      // E[f_j[d]^2]
            mu[d] = f[d] - T;
            float eg2 = f[d] * f[d] - 2.0f * f[d] * T + U;   // E[(f_i - f_j)^2]
            sg[d] = fast_sqrt(fmaxf(eg2 - mu[d] * mu[d], 0.0f) + 1e-6f);
        }
        float4* o = (float4*)(out + (size_t)ii * 12);
        o[0] = make_float4(mu[0], mu[1], mu[2], mu[3]);
        o[1] = make_float4(sg[0], sg[1], sg[2], sg[3]);
        o[2] = make_float4(gvx, gvy, f[2] - gvx, f[3] - gvy);
    }
}

// ---------------------------------------------------------------------------
// Launcher.  ws layout: [ rec: N*16B ][ G: N*64B ][ gv: 8B ]  (~320KB @ N=4096)
// ---------------------------------------------------------------------------
extern "C" void kernel_launch(void* const* d_in, const int* in_sizes, int n_in,
                              void* d_out, int out_size, void* d_ws, size_t ws_size,
                              hipStream_t stream) {
    const float* states  = (const float*)d_in[0];
    const float* log_tau = (const float*)d_in[1];
    const int N = in_sizes[0] / 8;           // 4096

    char* ws    = (char*)d_ws;
    float4* rec = (float4*)ws;
    float*  G   = (float*)(ws + (size_t)N * 16);
    float*  gv  = (float*)(ws + (size_t)N * 16 + (size_t)N * 64);
    float*  out = (float*)d_out;

    precompute_kernel<<<(N + 255) / 256, 256, 0, stream>>>(states, rec, G, N);
    groupvel_kernel<<<1, 256, 0, stream>>>(G, gv, N);
    neighbour_kernel<<<N / 128, 256, 0, stream>>>(rec, G, gv, log_tau, out, N);
}